// TimeSeriesSensorPredictor_33775622815752
// MI455X (gfx1250) — compile-verified
//
#include <hip/hip_runtime.h>
#include <math.h>

// ---- problem constants ----
#define B_    1024
#define S_IN_ 256
#define S_PR_ 128
#define E_    3
#define HID_  512
#define PE_CH_ 256.0f

typedef float v2f __attribute__((ext_vector_type(2)));
typedef float v8f __attribute__((ext_vector_type(8)));

// D = A(16x4 f32) * B(4x16 f32) + C(16x16 f32), wave32 WMMA.
// args: (neg_a, A, neg_b, B, c_mod, C, reuse_a, reuse_b)
static __device__ __forceinline__ v8f wmma4(v2f a, v2f b, v8f c) {
  return __builtin_amdgcn_wmma_f32_16x16x4_f32(false, a, false, b, (short)0, c,
                                               false, false);
}

static __device__ __forceinline__ float pe_val(int pos, int e) {
  // pos_enc with channels=256, out_ch=3 -> all three columns come from sin()
  float invf = powf(10000.0f, -(2.0f * (float)e) / PE_CH_);
  return sinf((float)pos * invf);
}

// ---------------- kernel 1: p = x + positional encoding ----------------
__global__ void pe_add_kernel(const float* __restrict__ x,
                              float* __restrict__ out, int S, int total) {
  int t = blockIdx.x * blockDim.x + threadIdx.x;
  if (t >= total) return;
  int pos = t % S;
#pragma unroll
  for (int e = 0; e < 3; ++e)
    out[(size_t)t * 3 + e] = x[(size_t)t * 3 + e] + pe_val(pos, e);
}

// ---------------- kernel 2: fused QKV projection (3x3 each) ----------------
// in_w: (9,3) row-major. rows 0-2 = Wq, 3-5 = Wk, 6-8 = Wv. in_b: (9)
__global__ void qkv_kernel(const float* __restrict__ srcq,
                           const float* __restrict__ srckv,
                           const float* __restrict__ w,
                           const float* __restrict__ bias,
                           float* __restrict__ q, float* __restrict__ k,
                           float* __restrict__ v, int nq, int nkv) {
  int t = blockIdx.x * blockDim.x + threadIdx.x;
  if (t < nq) {
    float x0 = srcq[(size_t)t * 3], x1 = srcq[(size_t)t * 3 + 1],
          x2 = srcq[(size_t)t * 3 + 2];
#pragma unroll
    for (int r = 0; r < 3; ++r)
      q[(size_t)t * 3 + r] =
          w[r * 3 + 0] * x0 + w[r * 3 + 1] * x1 + w[r * 3 + 2] * x2 + bias[r];
  }
  if (t < nkv) {
    float x0 = srckv[(size_t)t * 3], x1 = srckv[(size_t)t * 3 + 1],
          x2 = srckv[(size_t)t * 3 + 2];
#pragma unroll
    for (int r = 0; r < 3; ++r) {
      k[(size_t)t * 3 + r] = w[(3 + r) * 3 + 0] * x0 + w[(3 + r) * 3 + 1] * x1 +
                             w[(3 + r) * 3 + 2] * x2 + bias[3 + r];
      v[(size_t)t * 3 + r] = w[(6 + r) * 3 + 0] * x0 + w[(6 + r) * 3 + 1] * x1 +
                             w[(6 + r) * 3 + 2] * x2 + bias[6 + r];
    }
  }
}

// ---------------- kernel 3: attention (one wave per 16-query tile) ---------
// scores = (Q K^T)/sqrt(3) (+1.0 causal-tril mask), softmax, O = P V,
// out = O @ Wo^T + bo + resid.  LDS: 16 x SK score matrix (>= 256 floats).
__global__ __launch_bounds__(32) void attn_kernel(
    const float* __restrict__ q, const float* __restrict__ k,
    const float* __restrict__ v, const float* __restrict__ wo,
    const float* __restrict__ bo, const float* __restrict__ resid,
    float* __restrict__ out, int SQ, int SK, int causal) {
  extern __shared__ float lds[];  // 16*SK floats
  const int b = blockIdx.y;
  const int qt = blockIdx.x;
  const int lane = threadIdx.x;
  const int half = lane >> 4;   // 0: K=0,1 ; 1: K=2,3
  const int lidx = lane & 15;

  const float* qb = q + ((size_t)b * SQ + (size_t)qt * 16) * 3;
  const float* kb = k + (size_t)b * SK * 3;
  const float* vb = v + (size_t)b * SK * 3;
  const float scale = 0.5773502691896258f;  // 1/sqrt(E=3)

  // A fragment: Q tile 16x4 (K padded 3->4)
  v2f aq;
  aq.x = (half == 0) ? qb[lidx * 3 + 0] : qb[lidx * 3 + 2];
  aq.y = (half == 0) ? qb[lidx * 3 + 1] : 0.0f;

  const int nkt = SK >> 4;
  for (int kt = 0; kt < nkt; ++kt) {
    // B fragment: K^T tile 4x16 (rows = embed dim, cols = 16 keys)
    const float* kr = kb + (size_t)(kt * 16 + lidx) * 3;
    v2f bk;
    bk.x = (half == 0) ? kr[0] : kr[2];
    bk.y = (half == 0) ? kr[1] : 0.0f;
    v8f c;
#pragma unroll
    for (int r = 0; r < 8; ++r) c[r] = 0.0f;
    v8f d = wmma4(aq, bk, c);
#pragma unroll
    for (int r = 0; r < 8; ++r) {
      int row = r + 8 * half;            // query within tile
      int col = kt * 16 + lidx;          // key index
      float s = d[r] * scale;
      if (causal) s += (col <= qt * 16 + row) ? 1.0f : 0.0f;  // tril of ones
      lds[row * SK + col] = s;
    }
  }
  __syncthreads();

  // softmax: one row per lane (lanes 0..15)
  if (lane < 16) {
    float* row = lds + lane * SK;
    float mx = -3.0e38f;
    for (int j = 0; j < SK; ++j) mx = fmaxf(mx, row[j]);
    float sum = 0.0f;
    for (int j = 0; j < SK; ++j) {
      float e = __expf(row[j] - mx);
      row[j] = e;
      sum += e;
    }
    float inv = 1.0f / sum;
    for (int j = 0; j < SK; ++j) row[j] *= inv;
  }
  __syncthreads();

  // O = P (16 x SK) @ V (SK x 3), chained K=4 WMMAs
  v8f acc;
#pragma unroll
  for (int r = 0; r < 8; ++r) acc[r] = 0.0f;
  for (int kt = 0; kt < nkt; ++kt) {
#pragma unroll
    for (int kk = 0; kk < 4; ++kk) {
      int kbase = kt * 16 + kk * 4;
      v2f ap;  // A from probabilities in LDS
      ap.x = lds[lidx * SK + kbase + 2 * half + 0];
      ap.y = lds[lidx * SK + kbase + 2 * half + 1];
      v2f bv;  // B from V rows (N = value dim, padded 3->16)
      const float* vr = vb + (size_t)(kbase + 2 * half) * 3;
      bv.x = (lidx < 3) ? vr[lidx] : 0.0f;
      bv.y = (lidx < 3) ? vr[3 + lidx] : 0.0f;
      acc = wmma4(ap, bv, acc);
    }
  }
  __syncthreads();
  // stash O tile (16x16, cols 0..2 valid) into LDS for row-wise out-proj
#pragma unroll
  for (int r = 0; r < 8; ++r) lds[(r + 8 * half) * 16 + lidx] = acc[r];
  __syncthreads();

  if (lane < 16) {
    int tok = qt * 16 + lane;
    size_t gi = ((size_t)b * SQ + tok) * 3;
    float o0 = lds[lane * 16 + 0];
    float o1 = lds[lane * 16 + 1];
    float o2 = lds[lane * 16 + 2];
#pragma unroll
    for (int e = 0; e < 3; ++e) {
      out[gi + e] = wo[e * 3 + 0] * o0 + wo[e * 3 + 1] * o1 +
                    wo[e * 3 + 2] * o2 + bo[e] + resid[gi + e];
    }
  }
}

// ---------------- kernel 4: LayerNorm over (S, E) per batch sample --------
__global__ __launch_bounds__(256) void ln_kernel(const float* __restrict__ x,
                                                 const float* __restrict__ w,
                                                 const float* __restrict__ b,
                                                 float* __restrict__ out,
                                                 int S) {
  const int bidx = blockIdx.x;
  const int tid = threadIdx.x;
  const int N = S * 3;
  const float* xb = x + (size_t)bidx * N;
  __shared__ float rs_[256], rq_[256];
  float s = 0.0f, s2 = 0.0f;
  for (int i = tid; i < N; i += 256) {
    float t = xb[i];
    s += t;
    s2 += t * t;
  }
  rs_[tid] = s;
  rq_[tid] = s2;
  __syncthreads();
  for (int st = 128; st > 0; st >>= 1) {
    if (tid < st) {
      rs_[tid] += rs_[tid + st];
      rq_[tid] += rq_[tid + st];
    }
    __syncthreads();
  }
  float mean = rs_[0] / (float)N;
  float var = rq_[0] / (float)N - mean * mean;
  float rstd = rsqrtf(var + 1e-5f);
  for (int i = tid; i < N; i += 256)
    out[(size_t)bidx * N + i] = (xb[i] - mean) * rstd * w[i] + b[i];
}

// ---------------- kernel 5: FFN  out = relu(x W1^T + b1) W2^T + b2 + x ----
// one wave per 16-token tile; hidden 512 processed in 32 chunks of 16.
__global__ __launch_bounds__(32) void ffn_kernel(
    const float* __restrict__ x, const float* __restrict__ w1,
    const float* __restrict__ b1, const float* __restrict__ w2,
    const float* __restrict__ b2, float* __restrict__ out) {
  __shared__ float hl[16 * 16];
  const int tile = blockIdx.x;
  const int lane = threadIdx.x;
  const int half = lane >> 4;
  const int lidx = lane & 15;
  const float* xt = x + (size_t)tile * 16 * 3;

  // A fragment: 16 tokens x 4 (K padded 3->4)
  v2f a;
  a.x = (half == 0) ? xt[lidx * 3 + 0] : xt[lidx * 3 + 2];
  a.y = (half == 0) ? xt[lidx * 3 + 1] : 0.0f;

  v8f acc;  // output tile accumulator, init with b2 (N = out dim)
  float bias2 = (lidx < 3) ? b2[lidx] : 0.0f;
#pragma unroll
  for (int r = 0; r < 8; ++r) acc[r] = bias2;

  for (int j = 0; j < 32; ++j) {
    // GEMM1: hidden chunk = x @ W1^T[:, j*16..] + b1, via one WMMA (C = bias)
    int h = j * 16 + lidx;
    v2f b1f;
    b1f.x = (half == 0) ? w1[h * 3 + 0] : w1[h * 3 + 2];
    b1f.y = (half == 0) ? w1[h * 3 + 1] : 0.0f;
    v8f c1;
    float bb = b1[h];
#pragma unroll
    for (int r = 0; r < 8; ++r) c1[r] = bb;
    v8f hf = wmma4(a, b1f, c1);
#pragma unroll
    for (int r = 0; r < 8; ++r) hf[r] = fmaxf(hf[r], 0.0f);  // ReLU

    __syncthreads();  // protect previous iteration's LDS reads
#pragma unroll
    for (int r = 0; r < 8; ++r) hl[(r + 8 * half) * 16 + lidx] = hf[r];
    __syncthreads();

    // GEMM2: acc += hidden_chunk(16x16) @ W2^T chunk, 4 chained K=4 WMMAs
#pragma unroll
    for (int kk = 0; kk < 4; ++kk) {
      int kc = kk * 4 + 2 * half;
      v2f a2;
      a2.x = hl[lidx * 16 + kc + 0];
      a2.y = hl[lidx * 16 + kc + 1];
      int kg = j * 16 + kk * 4 + 2 * half;
      v2f b2f;  // W2 is (3,512) row-major; B[k][n] = w2[n*512 + k]
      b2f.x = (lidx < 3) ? w2[lidx * 512 + kg + 0] : 0.0f;
      b2f.y = (lidx < 3) ? w2[lidx * 512 + kg + 1] : 0.0f;
      acc = wmma4(a2, b2f, acc);
    }
  }

  // store (cols 0..2) + residual
  if (lidx < 3) {
#pragma unroll
    for (int r = 0; r < 8; ++r) {
      size_t tok = (size_t)tile * 16 + r + 8 * half;
      out[tok * 3 + lidx] = acc[r] + x[tok * 3 + lidx];
    }
  }
}

// ---------------- host orchestration ----------------
extern "C" void kernel_launch(void* const* d_in, const int* in_sizes, int n_in,
                              void* d_out, int out_size, void* d_ws,
                              size_t ws_size, hipStream_t stream) {
  (void)in_sizes; (void)n_in; (void)out_size; (void)ws_size;
  const float* X         = (const float*)d_in[0];
  const float* T         = (const float*)d_in[1];
  const float* enc_in_w  = (const float*)d_in[2];
  const float* enc_in_b  = (const float*)d_in[3];
  const float* enc_out_w = (const float*)d_in[4];
  const float* enc_out_b = (const float*)d_in[5];
  const float* enc_ln1_w = (const float*)d_in[6];
  const float* enc_ln1_b = (const float*)d_in[7];
  const float* enc_lin1_w= (const float*)d_in[8];
  const float* enc_lin1_b= (const float*)d_in[9];
  const float* enc_lin2_w= (const float*)d_in[10];
  const float* enc_lin2_b= (const float*)d_in[11];
  const float* enc_ln2_w = (const float*)d_in[12];
  const float* enc_ln2_b = (const float*)d_in[13];
  const float* dec_a1_in_w  = (const float*)d_in[14];
  const float* dec_a1_in_b  = (const float*)d_in[15];
  const float* dec_a1_out_w = (const float*)d_in[16];
  const float* dec_a1_out_b = (const float*)d_in[17];
  const float* dec_ln1_w = (const float*)d_in[18];
  const float* dec_ln1_b = (const float*)d_in[19];
  const float* dec_a2_in_w  = (const float*)d_in[20];
  const float* dec_a2_in_b  = (const float*)d_in[21];
  const float* dec_a2_out_w = (const float*)d_in[22];
  const float* dec_a2_out_b = (const float*)d_in[23];
  const float* dec_lin1_w= (const float*)d_in[24];
  const float* dec_lin1_b= (const float*)d_in[25];
  const float* dec_lin2_w= (const float*)d_in[26];
  const float* dec_lin2_b= (const float*)d_in[27];
  const float* dec_ln3_w = (const float*)d_in[28];
  const float* dec_ln3_b = (const float*)d_in[29];
  float* outp = (float*)d_out;

  const int ENC_TOK = B_ * S_IN_;            // 262144
  const int DEC_TOK = B_ * S_PR_;            // 131072
  const size_t ENC_ELEMS = (size_t)ENC_TOK * 3;

  float* ws = (float*)d_ws;
  float* pX     = ws;
  float* qb     = pX + ENC_ELEMS;
  float* kb     = qb + ENC_ELEMS;
  float* vb     = kb + ENC_ELEMS;
  float* sA     = vb + ENC_ELEMS;
  float* sB     = sA + ENC_ELEMS;
  float* encout = sB + ENC_ELEMS;
  float* pT     = encout + ENC_ELEMS;

  // ======== encoder ========
  pe_add_kernel<<<(ENC_TOK + 255) / 256, 256, 0, stream>>>(X, pX, S_IN_,
                                                           ENC_TOK);
  qkv_kernel<<<(ENC_TOK + 255) / 256, 256, 0, stream>>>(
      pX, pX, enc_in_w, enc_in_b, qb, kb, vb, ENC_TOK, ENC_TOK);
  {
    dim3 g(S_IN_ / 16, B_);
    attn_kernel<<<g, 32, 16 * S_IN_ * sizeof(float), stream>>>(
        qb, kb, vb, enc_out_w, enc_out_b, pX, sA, S_IN_, S_IN_, 0);
  }
  ln_kernel<<<B_, 256, 0, stream>>>(sA, enc_ln1_w, enc_ln1_b, sB, S_IN_);
  ffn_kernel<<<ENC_TOK / 16, 32, 0, stream>>>(sB, enc_lin1_w, enc_lin1_b,
                                              enc_lin2_w, enc_lin2_b, sA);
  ln_kernel<<<B_, 256, 0, stream>>>(sA, enc_ln2_w, enc_ln2_b, encout, S_IN_);

  // ======== decoder ========
  pe_add_kernel<<<(DEC_TOK + 255) / 256, 256, 0, stream>>>(T, pT, S_PR_,
                                                           DEC_TOK);
  // self attention (causal +1 tril mask, per reference)
  qkv_kernel<<<(DEC_TOK + 255) / 256, 256, 0, stream>>>(
      pT, pT, dec_a1_in_w, dec_a1_in_b, qb, kb, vb, DEC_TOK, DEC_TOK);
  {
    dim3 g(S_PR_ / 16, B_);
    attn_kernel<<<g, 32, 16 * S_PR_ * sizeof(float), stream>>>(
        qb, kb, vb, dec_a1_out_w, dec_a1_out_b, pT, sA, S_PR_, S_PR_, 1);
  }
  ln_kernel<<<B_, 256, 0, stream>>>(sA, dec_ln1_w, dec_ln1_b, sB, S_PR_); // x2
  // cross attention: Q from x2 (sB), K/V from encoder output
  qkv_kernel<<<(ENC_TOK + 255) / 256, 256, 0, stream>>>(
      sB, encout, dec_a2_in_w, dec_a2_in_b, qb, kb, vb, DEC_TOK, ENC_TOK);
  {
    dim3 g(S_PR_ / 16, B_);
    attn_kernel<<<g, 32, 16 * S_IN_ * sizeof(float), stream>>>(
        qb, kb, vb, dec_a2_out_w, dec_a2_out_b, sB, sA, S_PR_, S_IN_, 0);
  }
  ln_kernel<<<B_, 256, 0, stream>>>(sA, dec_ln1_w, dec_ln1_b, sB, S_PR_); // x3
  ffn_kernel<<<DEC_TOK / 16, 32, 0, stream>>>(sB, dec_lin1_w, dec_lin1_b,
                                              dec_lin2_w, dec_lin2_b, sA);
  ln_kernel<<<B_, 256, 0, stream>>>(sA, dec_ln3_w, dec_ln3_b, outp, S_PR_);
}